// FNOStage1_57028575756962
// MI455X (gfx1250) — compile-verified
//
#include <hip/hip_runtime.h>
#include <math.h>

// ---------------------------------------------------------------------------
// FNO stage for MI455X (gfx1250, wave32, WMMA).
// All dense math runs on V_WMMA_F32_16X16X4_F32 (f32 in / f32 acc): the stage
// is HBM-bound (AI ~16-27 flop/B vs ~26 flop/B machine balance), so full-f32
// matrix ops are free and keep reference accuracy.
// The truncated rfft2/irfft2 is expressed as 6 dense GEMMs against
// precomputed DFT twiddle tables (only 2*32 x 32 modes are retained).
// Each wave owns a 64(M) x 16(N) strip: 4 accumulators share one B-operand
// load pair -> 4 WMMAs per B fetch. The GEMM layout constants are template
// parameters so all inner-loop addressing strength-reduces to constant adds
// (no v_mul_u64 in the hot loop).
// Workspace requirement: 60,981,248 floats (~233 MiB).
// ---------------------------------------------------------------------------

typedef __attribute__((ext_vector_type(2))) float v2f;
typedef __attribute__((ext_vector_type(8))) float v8f;

#define WMMA_F32(a, b, c) \
  __builtin_amdgcn_wmma_f32_16x16x4_f32(false, (a), false, (b), (short)0, (c), false, false)

__device__ __forceinline__ float gelu_f(float v) {
  // exact erf form (torch nn.GELU default / jax approximate=False)
  return 0.5f * v * (1.0f + erff(v * 0.70710678118654752f));
}

// ---------------- workspace layout (float offsets) -------------------------
#define OFF_FW   0LL          // [64][256]   fwd W-DFT rows (m=2*ky+reim)
#define OFF_BRE  16384LL      // [64][512]   fwd H-DFT (re out), stacked [re|im] K
#define OFF_BIM  49152LL      // [64][512]   fwd H-DFT (im out)
#define OFF_HRE  81920LL      // [256][128]  inv H-DFT (re out)
#define OFF_HIM  114688LL     // [256][128]  inv H-DFT (im out)
#define OFF_WI   147456LL     // [256][64]   inv W-rDFT incl. hermitian 2x and 1/(H*W)
#define OFF_XW   163840LL     // [bc=256][ky=32][reim=2][h=256]
#define OFF_XF   4358144LL    // [bc=256][reim=2][kx=64][ky=32]
#define OFF_XFO  5406720LL    // [bo=256][reim=2][kx=64][ky=32]
#define OFF_Y1   6455296LL    // [bo=256][reim=2][ky=32][h=256]
#define OFF_Y    10649600LL   // [b=4][o=64][h=256][w=256]
#define OFF_XA   27426816LL   // activation ping
#define OFF_XB   44204032LL   // activation pong
#define WS_FLOATS 60981248LL

// ---------------- DFT twiddle tables ----------------------------------------
__global__ __launch_bounds__(256) void init_tables_kernel(float* __restrict__ t) {
  const int idx = blockIdx.x * blockDim.x + threadIdx.x;
  const float W0 = 6.28318530717958647692f / 256.0f;
  if (idx < 16384) {                       // Fw: forward rDFT along W
    const int m = idx >> 8, w = idx & 255;
    const int ky = m >> 1;
    const float th = W0 * (float)((ky * w) & 255);
    t[idx] = (m & 1) ? -sinf(th) : cosf(th);      // e^{-i th}: (cos, -sin)
  } else if (idx < 81920) {                // Bf_re / Bf_im: forward DFT along H
    int r = idx - 16384;
    const int which = r >> 15;             // 0 = re-out, 1 = im-out
    r &= 32767;
    const int kx = r >> 9, k = r & 511;
    const int kxv = (kx < 32) ? kx : kx + 192;    // retained rows 0..31, 224..255
    const int h = k & 255;
    const float th = W0 * (float)((kxv * h) & 255);
    const float c = cosf(th), s = sinf(th);
    // re = a*c + b*s ; im = b*c - a*s  over stacked K = [a(re) | b(im)]
    t[idx] = (which == 0) ? ((k < 256) ? c : s) : ((k < 256) ? -s : c);
  } else if (idx < 147456) {               // Hi_re / Hi_im: inverse DFT along H
    int r = idx - 81920;
    const int which = r >> 15;
    r &= 32767;
    const int h = r >> 7, k = r & 127;
    const int j = k & 63;
    const int kxv = (j < 32) ? j : j + 192;
    const float th = W0 * (float)((kxv * h) & 255);
    const float c = cosf(th), s = sinf(th);
    // e^{+i th}: re = a*c - b*s ; im = a*s + b*c  over stacked K = [re | im]
    t[idx] = (which == 0) ? ((k < 64) ? c : -s) : ((k < 64) ? s : c);
  } else if (idx < 163840) {               // Wi: inverse rDFT along W
    const int r = idx - 147456;
    const int w = r >> 6, k = r & 63;
    const int ky = k & 31;
    const float f = ((ky == 0) ? 1.0f : 2.0f) * (1.0f / 65536.0f);
    const float th = W0 * (float)((ky * w) & 255);
    t[idx] = (k < 32) ? f * cosf(th) : -f * sinf(th);
  }
}

// ---------------- generic f32-WMMA GEMM (wave = 64x16 D strip) -------------
// All layout constants are template parameters so the K-loop addressing is
// pure constant-increment induction variables.
//   B addr(k,n) = (n/NPERB)*BBATCH + (n%NPERB)*BN + k*BK
//   D addr(m,n) = (n/NPERD)*DBATCH + (n%NPERD)*DN + m*DM
template <int KK, int MGROUPS,
          long long BK, long long BN, long long BBATCH, int NPERB,
          long long DM, long long DN, long long DBATCH, int NPERD>
__global__ __launch_bounds__(256) void spec_gemm_kernel(
    const float* __restrict__ A, const float* __restrict__ B,
    float* __restrict__ D, int nTiles) {
  const int wave = threadIdx.x >> 5;
  const int lane = threadIdx.x & 31;
  const long long t = (long long)blockIdx.x * 8 + wave;
  if (t >= (long long)MGROUPS * nTiles) return;
  const int mg = (int)(t % MGROUPS);
  const int ntile = (int)(t / MGROUPS);
  const int l15 = lane & 15;
  const int hi  = lane >> 4;
  const int khalf = hi << 1;
  const int n = (ntile << 4) + l15;
  const float* Ap = A + (long long)((mg << 6) + l15) * KK;
  const float* Bp = B + (long long)(n / NPERB) * BBATCH +
                        (long long)(n % NPERB) * BN;
  v8f acc0 = {}, acc1 = {}, acc2 = {}, acc3 = {};
  for (int k = 0; k < KK; k += 4) {
    const int kb = k + khalf;
    v2f b;
    b.x = Bp[(long long)kb * BK];
    b.y = Bp[(long long)(kb + 1) * BK];
    if (k + 16 < KK) __builtin_prefetch(Bp + (long long)(kb + 16) * BK, 0, 3);
    v2f a;
    a.x = Ap[kb];                  a.y = Ap[kb + 1];
    acc0 = WMMA_F32(a, b, acc0);
    a.x = Ap[1 * 16 * KK + kb];    a.y = Ap[1 * 16 * KK + kb + 1];
    acc1 = WMMA_F32(a, b, acc1);
    a.x = Ap[2 * 16 * KK + kb];    a.y = Ap[2 * 16 * KK + kb + 1];
    acc2 = WMMA_F32(a, b, acc2);
    a.x = Ap[3 * 16 * KK + kb];    a.y = Ap[3 * 16 * KK + kb + 1];
    acc3 = WMMA_F32(a, b, acc3);
  }
  float* Dp = D + (long long)(n / NPERD) * DBATCH +
                  (long long)(n % NPERD) * DN;
#pragma unroll
  for (int mt = 0; mt < 4; ++mt) {
    const v8f acc = (mt == 0) ? acc0 : (mt == 1) ? acc1 : (mt == 2) ? acc2 : acc3;
#pragma unroll
    for (int r = 0; r < 8; ++r) {
      const int mm = (mg << 6) + (mt << 4) + r + (hi << 3);
      Dp[(long long)mm * DM] = acc[r];
    }
  }
}

// ---------------- per-mode complex channel mix (tiny, weight-BW bound) -----
__global__ __launch_bounds__(256) void spec_mul_kernel(
    const float* __restrict__ XF, const float* __restrict__ sw1,
    const float* __restrict__ sw2, float* __restrict__ XFo) {
  __shared__ float xs[4][64][2];
  const int kx = blockIdx.x >> 5;
  const int ky = blockIdx.x & 31;
  const int tid = threadIdx.x;
#pragma unroll
  for (int i = 0; i < 2; ++i) {
    const int idx = (i << 8) + tid;          // 0..511 over (b, ch, reim)
    const int bb = idx >> 7;
    const int ch = (idx >> 1) & 63;
    const int ri = idx & 1;
    xs[bb][ch][ri] =
        XF[(long long)(bb * 64 + ch) * 4096 + ri * 2048 + kx * 32 + ky];
  }
  __syncthreads();
  const int b = tid >> 6;
  const int o = tid & 63;
  const float* w = (kx < 32) ? sw1 : sw2;
  const int m1 = (kx < 32) ? kx : kx - 32;
  float ar = 0.0f, ai = 0.0f;
  for (int i = 0; i < 64; ++i) {
    const long long wb = ((((long long)i * 64 + o) * 32 + m1) * 32 + ky) * 2;
    const float wr = w[wb], wi = w[wb + 1];
    const float xr = xs[b][i][0], xi = xs[b][i][1];
    ar = fmaf(xr, wr, fmaf(-xi, wi, ar));
    ai = fmaf(xr, wi, fmaf(xi, wr, ai));
  }
  const long long ob = (long long)(b * 64 + o) * 4096 + kx * 32 + ky;
  XFo[ob] = ar;
  XFo[ob + 2048] = ai;
}

// ---------------- skip conv fused with spectral add + GELU -----------------
// One wave covers all 64 output channels of a 16-pixel column: x tile is
// loaded from global exactly once.
__global__ __launch_bounds__(256) void chan_skip_kernel(
    const float* __restrict__ x, const float* __restrict__ y,
    const float* __restrict__ W, const float* __restrict__ bias,
    float* __restrict__ out) {
  const int HW = 65536;
  const int wave = threadIdx.x >> 5;
  const int lane = threadIdx.x & 31;
  const long long t = (long long)blockIdx.x * 8 + wave;  // 16384 n-tiles
  if (t >= 16384) return;
  const int l15 = lane & 15, hi = lane >> 4, khalf = hi << 1;
  const int n = (int)(t << 4) + l15;
  const int b = n >> 16, hw = n & 65535;
  const float* Bp = x + (long long)b * 64 * HW + hw;
  const float* Ap = W + l15 * 64;
  v8f acc0 = {}, acc1 = {}, acc2 = {}, acc3 = {};
#pragma unroll
  for (int k = 0; k < 64; k += 4) {
    const int kb = k + khalf;
    v2f bb;
    bb.x = Bp[(long long)kb * HW];
    bb.y = Bp[(long long)(kb + 1) * HW];
    v2f a;
    a.x = Ap[kb];         a.y = Ap[kb + 1];
    acc0 = WMMA_F32(a, bb, acc0);
    a.x = Ap[1024 + kb];  a.y = Ap[1024 + kb + 1];
    acc1 = WMMA_F32(a, bb, acc1);
    a.x = Ap[2048 + kb];  a.y = Ap[2048 + kb + 1];
    acc2 = WMMA_F32(a, bb, acc2);
    a.x = Ap[3072 + kb];  a.y = Ap[3072 + kb + 1];
    acc3 = WMMA_F32(a, bb, acc3);
  }
  const long long obase = (long long)b * 64 * HW + hw;
#pragma unroll
  for (int mt = 0; mt < 4; ++mt) {
    const v8f acc = (mt == 0) ? acc0 : (mt == 1) ? acc1 : (mt == 2) ? acc2 : acc3;
#pragma unroll
    for (int r = 0; r < 8; ++r) {
      const int mm = (mt << 4) + r + (hi << 3);
      const long long o = obase + (long long)mm * HW;
      out[o] = gelu_f(acc[r] + bias[mm] + y[o]);
    }
  }
}

// ---------------- fused channel MLP: fc1->GELU->(fc2 + skip)->GELU ---------
// Hidden tile (128 x 16) lives in LDS per wave: saves a 128 MB round-trip.
__global__ __launch_bounds__(128) void mlp_fused_kernel(
    const float* __restrict__ x,
    const float* __restrict__ W1, const float* __restrict__ b1,
    const float* __restrict__ W2, const float* __restrict__ b2,
    const float* __restrict__ Ws, const float* __restrict__ bs,
    float* __restrict__ out) {
  __shared__ float sx[4][64 * 16];
  __shared__ float sh[4][128 * 16];
  const int HW = 65536;
  const int wave = threadIdx.x >> 5;
  const int lane = threadIdx.x & 31;
  const int l15 = lane & 15, hi = lane >> 4, khalf = hi << 1;
  const long long nt = (long long)blockIdx.x * 4 + wave;   // 16384 n-tiles
  const int n0 = (int)(nt << 4);
  const int b = n0 >> 16, hw0 = n0 & 65535;
  const float* xbase = x + (long long)b * 64 * HW + hw0;
  float* xs = sx[wave];
  float* hs = sh[wave];
  // stage x tile [64 x 16] to LDS (single global read, reused by 2 GEMMs)
  for (int i = 0; i < 32; ++i) {
    const int kn = (i << 5) + lane;
    xs[kn] = xbase[(long long)(kn >> 4) * HW + (kn & 15)];
  }
  // fc1 + GELU -> LDS  (M = 128, K = 64), two groups of 4 m-tiles
  for (int g2 = 0; g2 < 2; ++g2) {
    const float* Ap = W1 + ((g2 << 6) + l15) * 64;
    v8f acc0 = {}, acc1 = {}, acc2 = {}, acc3 = {};
#pragma unroll
    for (int k = 0; k < 64; k += 4) {
      const int kb = k + khalf;
      v2f bb;
      bb.x = xs[(kb << 4) + l15];
      bb.y = xs[((kb + 1) << 4) + l15];
      v2f a;
      a.x = Ap[kb];         a.y = Ap[kb + 1];
      acc0 = WMMA_F32(a, bb, acc0);
      a.x = Ap[1024 + kb];  a.y = Ap[1024 + kb + 1];
      acc1 = WMMA_F32(a, bb, acc1);
      a.x = Ap[2048 + kb];  a.y = Ap[2048 + kb + 1];
      acc2 = WMMA_F32(a, bb, acc2);
      a.x = Ap[3072 + kb];  a.y = Ap[3072 + kb + 1];
      acc3 = WMMA_F32(a, bb, acc3);
    }
#pragma unroll
    for (int mt = 0; mt < 4; ++mt) {
      const v8f acc = (mt == 0) ? acc0 : (mt == 1) ? acc1 : (mt == 2) ? acc2 : acc3;
#pragma unroll
      for (int r = 0; r < 8; ++r) {
        const int mm = (g2 << 6) + (mt << 4) + r + (hi << 3);
        hs[(mm << 4) + l15] = gelu_f(acc[r] + b1[mm]);
      }
    }
  }
  // fc2 (K = 128 from LDS) + 1x1 skip (K = 64 from LDS) + GELU  (M = 64)
  {
    const float* A2 = W2 + l15 * 128;
    const float* As = Ws + l15 * 64;
    v8f acc0 = {}, acc1 = {}, acc2 = {}, acc3 = {};
#pragma unroll
    for (int k = 0; k < 128; k += 4) {
      const int kb = k + khalf;
      v2f bb;
      bb.x = hs[(kb << 4) + l15];
      bb.y = hs[((kb + 1) << 4) + l15];
      v2f a;
      a.x = A2[kb];         a.y = A2[kb + 1];
      acc0 = WMMA_F32(a, bb, acc0);
      a.x = A2[2048 + kb];  a.y = A2[2048 + kb + 1];
      acc1 = WMMA_F32(a, bb, acc1);
      a.x = A2[4096 + kb];  a.y = A2[4096 + kb + 1];
      acc2 = WMMA_F32(a, bb, acc2);
      a.x = A2[6144 + kb];  a.y = A2[6144 + kb + 1];
      acc3 = WMMA_F32(a, bb, acc3);
    }
#pragma unroll
    for (int k = 0; k < 64; k += 4) {
      const int kb = k + khalf;
      v2f bb;
      bb.x = xs[(kb << 4) + l15];
      bb.y = xs[((kb + 1) << 4) + l15];
      v2f a;
      a.x = As[kb];         a.y = As[kb + 1];
      acc0 = WMMA_F32(a, bb, acc0);
      a.x = As[1024 + kb];  a.y = As[1024 + kb + 1];
      acc1 = WMMA_F32(a, bb, acc1);
      a.x = As[2048 + kb];  a.y = As[2048 + kb + 1];
      acc2 = WMMA_F32(a, bb, acc2);
      a.x = As[3072 + kb];  a.y = As[3072 + kb + 1];
      acc3 = WMMA_F32(a, bb, acc3);
    }
#pragma unroll
    for (int mt = 0; mt < 4; ++mt) {
      const v8f acc = (mt == 0) ? acc0 : (mt == 1) ? acc1 : (mt == 2) ? acc2 : acc3;
#pragma unroll
      for (int r = 0; r < 8; ++r) {
        const int mm = (mt << 4) + r + (hi << 3);
        out[(long long)(b * 64 + mm) * HW + hw0 + l15] =
            gelu_f(acc[r] + b2[mm] + bs[mm]);
      }
    }
  }
}

// ---------------- fused projection: fc1->GELU->fc2 (O = 1) -----------------
// Hidden 256-ch vector never touches HBM (saves a 512 MB round-trip).
__global__ __launch_bounds__(256) void proj_fused_kernel(
    const float* __restrict__ x,
    const float* __restrict__ W1, const float* __restrict__ b1,
    const float* __restrict__ w2, const float* __restrict__ b2,
    float* __restrict__ out) {
  __shared__ float sx[8][64 * 16];
  const int HW = 65536;
  const int wave = threadIdx.x >> 5;
  const int lane = threadIdx.x & 31;
  const int l15 = lane & 15, hi = lane >> 4, khalf = hi << 1;
  const long long nt = (long long)blockIdx.x * 8 + wave;
  const int n0 = (int)(nt << 4);
  const int b = n0 >> 16, hw0 = n0 & 65535;
  const float* xbase = x + (long long)b * 64 * HW + hw0;
  float* xs = sx[wave];
  for (int i = 0; i < 32; ++i) {
    const int kn = (i << 5) + lane;
    xs[kn] = xbase[(long long)(kn >> 4) * HW + (kn & 15)];
  }
  float partial = 0.0f;
  for (int mg = 0; mg < 4; ++mg) {          // 4 groups of 4 m-tiles (M = 256)
    const float* Ap = W1 + ((mg << 6) + l15) * 64;
    v8f acc0 = {}, acc1 = {}, acc2 = {}, acc3 = {};
#pragma unroll
    for (int k = 0; k < 64; k += 4) {
      const int kb = k + khalf;
      v2f bb;
      bb.x = xs[(kb << 4) + l15];
      bb.y = xs[((kb + 1) << 4) + l15];
      v2f a;
      a.x = Ap[kb];         a.y = Ap[kb + 1];
      acc0 = WMMA_F32(a, bb, acc0);
      a.x = Ap[1024 + kb];  a.y = Ap[1024 + kb + 1];
      acc1 = WMMA_F32(a, bb, acc1);
      a.x = Ap[2048 + kb];  a.y = Ap[2048 + kb + 1];
      acc2 = WMMA_F32(a, bb, acc2);
      a.x = Ap[3072 + kb];  a.y = Ap[3072 + kb + 1];
      acc3 = WMMA_F32(a, bb, acc3);
    }
#pragma unroll
    for (int mt = 0; mt < 4; ++mt) {
      const v8f acc = (mt == 0) ? acc0 : (mt == 1) ? acc1 : (mt == 2) ? acc2 : acc3;
#pragma unroll
      for (int r = 0; r < 8; ++r) {
        const int mm = (mg << 6) + (mt << 4) + r + (hi << 3);
        partial += w2[mm] * gelu_f(acc[r] + b1[mm]);
      }
    }
  }
  partial += __shfl_xor(partial, 16, 32);   // combine m-halves across lanes
  if (hi == 0) out[(long long)b * HW + hw0 + l15] = partial + b2[0];
}

// ---------------------------------------------------------------------------
extern "C" void kernel_launch(void* const* d_in, const int* in_sizes, int n_in,
                              void* d_out, int out_size, void* d_ws, size_t ws_size,
                              hipStream_t stream) {
  const float* x_in       = (const float*)d_in[0];
  const float* sw1        = (const float*)d_in[1];
  const float* sw2        = (const float*)d_in[2];
  const float* skip_w     = (const float*)d_in[3];
  const float* skip_b     = (const float*)d_in[4];
  const float* mlp_w1     = (const float*)d_in[5];
  const float* mlp_b1     = (const float*)d_in[6];
  const float* mlp_w2     = (const float*)d_in[7];
  const float* mlp_b2     = (const float*)d_in[8];
  const float* mlp_skip_w = (const float*)d_in[9];
  const float* mlp_skip_b = (const float*)d_in[10];
  const float* proj_w1    = (const float*)d_in[11];
  const float* proj_b1    = (const float*)d_in[12];
  const float* proj_w2    = (const float*)d_in[13];
  const float* proj_b2    = (const float*)d_in[14];
  (void)in_sizes; (void)n_in; (void)out_size; (void)ws_size;

  float* ws  = (float*)d_ws;
  float* tbl = ws;
  float* Xw  = ws + OFF_XW;
  float* XF  = ws + OFF_XF;
  float* XFo = ws + OFF_XFO;
  float* Y1  = ws + OFF_Y1;
  float* yb  = ws + OFF_Y;
  float* xA  = ws + OFF_XA;
  float* xB  = ws + OFF_XB;

  init_tables_kernel<<<640, 256, 0, stream>>>(tbl);

  const float* xcur = x_in;
  for (int l = 0; l < 2; ++l) {
    // A: truncated rDFT along W  (M=64 mode-reals, N=65536 rows, K=256)
    spec_gemm_kernel<256, 1, 1LL, 256LL, 0LL, (1 << 30),
                     256LL, 1LL, 16384LL, 256>
        <<<512, 256, 0, stream>>>(tbl + OFF_FW, xcur, Xw, 4096);
    // B: truncated DFT along H on complex data (re-out then im-out)
    spec_gemm_kernel<512, 1, 1LL, 512LL, 16384LL, 32,
                     32LL, 1LL, 4096LL, 32>
        <<<64, 256, 0, stream>>>(tbl + OFF_BRE, Xw, XF, 512);
    spec_gemm_kernel<512, 1, 1LL, 512LL, 16384LL, 32,
                     32LL, 1LL, 4096LL, 32>
        <<<64, 256, 0, stream>>>(tbl + OFF_BIM, Xw, XF + 2048, 512);
    // C: per-retained-mode complex channel mix (sw1: low rows, sw2: high rows)
    spec_mul_kernel<<<2048, 256, 0, stream>>>(
        XF, sw1 + (long long)l * 8388608, sw2 + (long long)l * 8388608, XFo);
    // D: inverse DFT along H (complex)
    spec_gemm_kernel<128, 4, 32LL, 1LL, 4096LL, 32,
                     1LL, 256LL, 16384LL, 32>
        <<<256, 256, 0, stream>>>(tbl + OFF_HRE, XFo, Y1, 512);
    spec_gemm_kernel<128, 4, 32LL, 1LL, 4096LL, 32,
                     1LL, 256LL, 16384LL, 32>
        <<<256, 256, 0, stream>>>(tbl + OFF_HIM, XFo, Y1 + 8192, 512);
    // E: inverse rDFT along W (hermitian 2x + 1/(H*W) folded into table)
    spec_gemm_kernel<64, 4, 256LL, 1LL, 16384LL, 256,
                     1LL, 256LL, 65536LL, 256>
        <<<2048, 256, 0, stream>>>(tbl + OFF_WI, Y1, yb, 4096);
    // pointwise skip + spectral add + GELU
    chan_skip_kernel<<<2048, 256, 0, stream>>>(
        xcur, yb, skip_w + l * 4096, skip_b + l * 64, xA);
    // fused channel MLP with 1x1 skip
    mlp_fused_kernel<<<4096, 128, 0, stream>>>(
        xA, mlp_w1 + l * 8192, mlp_b1 + l * 128,
        mlp_w2 + l * 8192, mlp_b2 + l * 64,
        mlp_skip_w + l * 4096, mlp_skip_b + l * 64, xB);
    xcur = xB;
  }
  // fused projection 64 -> 256 -> 1
  proj_fused_kernel<<<2048, 256, 0, stream>>>(
      xcur, proj_w1, proj_b1, proj_w2, proj_b2, (float*)d_out);
}